// LogSigProdLayer_33517924778479
// MI455X (gfx1250) — compile-verified
//
#include <hip/hip_runtime.h>
#include <hip/hip_bf16.h>
#include <stdint.h>
#include <math.h>

// Geometry fixed by the reference: x[8,4096], W[4096,4096] (95% zeros), bias[4096].
#define NIN            4096
#define NOUT           4096
#define BATCH          8
#define COLS           128              // columns per block
#define ROWS_PER_WAVE  32               // lane = row within the wave tile
#define NWAVES         8
#define ROWS_PER_BLOCK (ROWS_PER_WAVE * NWAVES)   // 256
#define NTHREADS       256
#define NRC            (NIN / ROWS_PER_WAVE)      // 128 row-chunks total
#define LN2_F          0.69314718055994531f

#define WSTRIDE 129                               // padded LDS row stride (floats)
#define WTILE_F (ROWS_PER_WAVE * WSTRIDE)         // 4128 floats per wave tile

typedef float v4f  __attribute__((ext_vector_type(4)));
typedef int   v4i  __attribute__((ext_vector_type(4)));
typedef unsigned int u32x4 __attribute__((ext_vector_type(4)));
typedef int   i32x8 __attribute__((ext_vector_type(8)));
typedef int   i32x4 __attribute__((ext_vector_type(4)));
typedef __attribute__((address_space(1))) v4i* gv4i_p;   // global (AS1)
typedef __attribute__((address_space(3))) v4i* lv4i_p;   // LDS    (AS3)

// ---- raw HW transcendentals (base-2): v_exp_f32 / v_log_f32, no OCML fixup ----
__device__ __forceinline__ float fast_exp2(float x) {
#if __has_builtin(__builtin_amdgcn_exp2f)
  return __builtin_amdgcn_exp2f(x);
#else
  return exp2f(x);   // host pass only parses this; device pass uses the builtin
#endif
}
__device__ __forceinline__ float fast_log2(float x) {
#if __has_builtin(__builtin_amdgcn_logf)
  return __builtin_amdgcn_logf(x);
#else
  return log2f(x);   // host pass only parses this; device pass uses the builtin
#endif
}

// ---- CDNA5 async global->LDS copy (verified to compile in round 2) ----
__device__ __forceinline__ void async_copy_b128(const void* gsrc, void* ldst) {
#if __has_builtin(__builtin_amdgcn_global_load_async_to_lds_b128)
  __builtin_amdgcn_global_load_async_to_lds_b128((gv4i_p)gsrc, (lv4i_p)ldst, 0, 0);
#else
  uint32_t lds_addr = (uint32_t)(uintptr_t)(lv4i_p)ldst;
  asm volatile("global_load_async_to_lds_b128 %0, %1, off"
               :: "v"(lds_addr), "v"((uint64_t)(uintptr_t)gsrc) : "memory");
#endif
}
__device__ __forceinline__ void wait_async0() {
#if __has_builtin(__builtin_amdgcn_s_wait_asynccnt)
  __builtin_amdgcn_s_wait_asynccnt(0);
#else
  asm volatile("s_wait_asynccnt 0" ::: "memory");
#endif
}
__device__ __forceinline__ void wait_tensor0() {
#if __has_builtin(__builtin_amdgcn_s_wait_tensorcnt)
  __builtin_amdgcn_s_wait_tensorcnt(0);
#else
  asm volatile("s_wait_tensorcnt 0" ::: "memory");
#endif
}

// ---- Kernel A: lx2[i][b] = log2(relu(x[b][i]) + 1e-3), i-major ----
__global__ __launch_bounds__(NTHREADS)
void lx_prep_kernel(const float* __restrict__ x, float* __restrict__ lxT) {
  int idx = blockIdx.x * NTHREADS + threadIdx.x;   // 0 .. 8*4096-1
  int i = idx >> 3;
  int b = idx & (BATCH - 1);
  float v = x[b * NIN + i];
  lxT[idx] = fast_log2(fmaxf(v, 0.0f) + 1.0e-3f);  // lxT[i*8 + b]
}

// ---- Kernel B: sparsity-compacted main kernel ----
// Wave-private tile: 32 rows x 128 cols. lane = row. Per column: ballot the
// nonzeros, process them 4-at-a-time across lanes (4 nz x 8 batches = 32 lanes).
__global__ __launch_bounds__(NTHREADS)
void logsig_sparse_kernel(const float* __restrict__ W,
                          const float* __restrict__ lxT,
                          float* __restrict__ pws) {
  __shared__ float wtile[NWAVES * WTILE_F];            // 8 x 16.5 KB padded tiles
  __shared__ float lxs[ROWS_PER_BLOCK * BATCH];        // 8 KB

  const int lane = threadIdx.x & 31;
  const int wv   = threadIdx.x >> 5;
  const int j0   = blockIdx.x * COLS;
  const int i0w  = blockIdx.y * ROWS_PER_BLOCK + wv * ROWS_PER_WAVE;
  const int rc   = blockIdx.y * NWAVES + wv;           // row-chunk id, 0..127

  // Stage this wave's lx rows (32 x 8 floats = 1 KB): async global->LDS.
  float* lxw = lxs + wv * (ROWS_PER_WAVE * BATCH);
  {
    const v4f* gsrc = (const v4f*)(lxT + (size_t)i0w * BATCH);  // 64 float4
    v4f* ldst = (v4f*)lxw;
    async_copy_b128(gsrc + lane,      ldst + lane);
    async_copy_b128(gsrc + lane + 32, ldst + lane + 32);
  }

  // Stage this wave's W tile via the Tensor Data Mover with LDS padding:
  // tile 128x32 f32, tensor row stride 4096, pad 1 DWORD every 128 DWORDs
  // -> LDS row stride 129 floats (bank-conflict-free column reads).
  float* wt = wtile + wv * WTILE_F;
#if __has_builtin(__builtin_amdgcn_tensor_load_to_lds)
  {
    uint64_t ga    = (uint64_t)(uintptr_t)(W + (size_t)i0w * NOUT + j0);
    uint32_t lbyte = (uint32_t)(uintptr_t)(__attribute__((address_space(3))) void*)wt;
    u32x4 g0;
    g0[0] = 1u;                                         // count=1 (valid), user mode
    g0[1] = lbyte;                                      // lds_addr
    g0[2] = (uint32_t)ga;                               // global_addr[31:0]
    g0[3] = (uint32_t)((ga >> 32) & 0x01ffffffu) | (2u << 30);  // addr[56:32], type=2
    i32x8 g1;
    g1[0] = (int)((2u << 16)        // data_size = 4 bytes
                | (1u << 20)        // pad_enable
                | (6u << 22)        // pad_interval: 128 DWORDs
                | (0u << 25));      // pad_amount:   1 DWORD
    g1[1] = (int)(((uint32_t)COLS          & 0xffffu) << 16);  // tensor_dim0[15:0]
    g1[2] = (int)(((uint32_t)ROWS_PER_WAVE & 0xffffu) << 16);  // tensor_dim1[15:0]
    g1[3] = (int)(((uint32_t)COLS          & 0xffffu) << 16);  // tile_dim0 = 128
    g1[4] = ROWS_PER_WAVE;                                     // tile_dim1 = 32
    g1[5] = NOUT;                                              // tensor_dim0_stride = 4096
    g1[6] = 0;
    g1[7] = 0;
    i32x4 g2 = {0, 0, 0, 0};                                   // 2D: groups 2/3 unused
    i32x4 g3 = {0, 0, 0, 0};
    i32x8 g4 = {0, 0, 0, 0, 0, 0, 0, 0};                       // VADDR4: unused
    __builtin_amdgcn_tensor_load_to_lds(g0, g1, g2, g3, g4, 0);
  }
  wait_tensor0();
#else
  for (int r = 0; r < ROWS_PER_WAVE; ++r) {            // fallback: cooperative copy
    v4f v = *(const v4f*)(W + (size_t)(i0w + r) * NOUT + j0 + lane * 4);
    wt[r * WSTRIDE + lane * 4 + 0] = v.x;
    wt[r * WSTRIDE + lane * 4 + 1] = v.y;
    wt[r * WSTRIDE + lane * 4 + 2] = v.z;
    wt[r * WSTRIDE + lane * 4 + 3] = v.w;
  }
#endif
  wait_async0();
  asm volatile("" ::: "memory");                       // order LDS reads after waits

  const float* wrow = wt + lane * WSTRIDE;             // this lane's row
  const int  b   = lane & 7;                           // batch this lane evaluates
  const bool gb0 = (lane & 8)  != 0;                   // nz-slot select bits
  const bool gb1 = (lane & 16) != 0;
  const int  g   = lane >> 3;                          // nz slot 0..3

#pragma unroll 2
  for (int col = 0; col < COLS; ++col) {
    float w = wrow[col];                               // conflict-free (stride 129)
    uint32_t m = (uint32_t)__ballot(w != 0.0f);        // ~1.6 of 32 set
    float res = 0.0f;
    while (m) {                                        // uniform loop, usually 1 iter
      int cnt = __builtin_popcount(m);
      uint32_t mm = m;
      int r0 = (int)__builtin_ctz(mm);                 mm &= mm - 1;
      int r1 = (int)__builtin_ctz(mm | 0x80000000u);   mm &= mm - 1;
      int r2 = (int)__builtin_ctz(mm | 0x80000000u);   mm &= mm - 1;
      int r3 = (int)__builtin_ctz(mm | 0x80000000u);   mm &= mm - 1;
      m = mm;
      int s01  = gb0 ? r1 : r0;
      int s23  = gb0 ? r3 : r2;
      int isel = gb1 ? s23 : s01;                      // row slot for this lane
      float wsel = __shfl(w, isel, 32);                // gather w from lane isel
      float lxv  = lxw[isel * BATCH + b];              // log2(relu(x)+eps)
      float z = wsel * lxv;                            // z = m/ln2
      float e = fast_exp2(-fabsf(z));                  // v_exp_f32 (neg-abs modifier)
      float t = fast_log2(1.0f + e);                   // v_log_f32
      float c = fminf(z, 0.0f) + 1.0f - t;             // (logsig(m)+log2)/ln2
      res += (g < cnt) ? c : 0.0f;
    }
    res += __shfl_xor(res, 16, 32);                    // fold 4 nz slots -> batch b
    res += __shfl_xor(res, 8, 32);
    if (lane < 8)                                      // lanes 0..7 hold batches 0..7
      pws[((size_t)rc * NOUT + (j0 + col)) * BATCH + lane] = res * LN2_F;
  }
}

// ---- Kernel C: fold the 128 row-chunks, add bias, scale ----
__global__ __launch_bounds__(NTHREADS)
void finish_kernel(const float* __restrict__ pws,
                   const float* __restrict__ bias,
                   float* __restrict__ out) {
  int idx = blockIdx.x * NTHREADS + threadIdx.x;       // b*4096 + j
  int b = idx >> 12;
  int j = idx & (NOUT - 1);
  float s = bias[j];
#pragma unroll 8
  for (int rcc = 0; rcc < NRC; ++rcc)
    s += pws[((size_t)rcc * NOUT + j) * BATCH + b];
  out[idx] = s * 1.0e-4f;
}

extern "C" void kernel_launch(void* const* d_in, const int* in_sizes, int n_in,
                              void* d_out, int out_size, void* d_ws, size_t ws_size,
                              hipStream_t stream) {
  const float* x    = (const float*)d_in[0];   // [8, 4096]
  const float* W    = (const float*)d_in[1];   // [4096, 4096]
  const float* bias = (const float*)d_in[2];   // [4096]
  float* out = (float*)d_out;                  // [8, 4096]

  float* lxT = (float*)d_ws;                   // 4096*8 floats (128 KB)
  float* pws = lxT + (size_t)NIN * BATCH;      // 128*4096*8 floats (16 MB)

  lx_prep_kernel<<<dim3((BATCH * NIN) / NTHREADS), NTHREADS, 0, stream>>>(x, lxT);
  logsig_sparse_kernel<<<dim3(NOUT / COLS, NIN / ROWS_PER_BLOCK), NTHREADS, 0, stream>>>(W, lxT, pws);
  finish_kernel<<<dim3((BATCH * NOUT) / NTHREADS), NTHREADS, 0, stream>>>(pws, bias, out);
}